// GCNLayer_48765058679293
// MI455X (gfx1250) — compile-verified
//
#include <hip/hip_runtime.h>
#include <cstdint>
#include <cstddef>

#define N_NODES 100000
#define N_EDGES 1600000
#define D 128
#define AGG_STRIDE 132   // 16 rows * 132 pitch: bank = (4*m + k) % 64 -> conflict-free A reads

typedef __attribute__((ext_vector_type(2))) float v2f;
typedef __attribute__((ext_vector_type(8))) float v8f;

// Hardware f32 atomic add (no return), device scope, saddr form:
//   global_atomic_add_f32 v_off, v_data, s[base:base+1] scope:SCOPE_DEV
__device__ __forceinline__ void gatomic_add_f32(const float* base, uint32_t byte_off, float v) {
    asm volatile("global_atomic_add_f32 %0, %1, %2 scope:SCOPE_DEV"
                 :: "v"(byte_off), "v"(v), "s"(base)
                 : "memory");
}

// ---------------------------------------------------------------------------
// Kernel 1: zero the workspace (sum[N*D] + deg[N]), float4 stores.
// ---------------------------------------------------------------------------
__global__ __launch_bounds__(256) void zero_ws_kernel(float4* __restrict__ ws, int n4) {
    int i = blockIdx.x * blockDim.x + threadIdx.x;
    int stride = gridDim.x * blockDim.x;
    float4 z = make_float4(0.f, 0.f, 0.f, 0.f);
    for (; i < n4; i += stride) ws[i] = z;
}

// ---------------------------------------------------------------------------
// Kernel 2: edge-parallel gather + scatter-add.
// One wave per edge iteration: 32 lanes x float4 = 512B coalesced row read,
// then 128 device-scope global_atomic_add_f32 into sum[dst], +1 into deg[dst].
// ---------------------------------------------------------------------------
__global__ __launch_bounds__(256) void edge_aggregate_kernel(
    const float* __restrict__ feature, const int* __restrict__ src,
    const int* __restrict__ dst, const float* __restrict__ sum,
    const float* __restrict__ deg) {
    const int lane = threadIdx.x & 31;
    const int wid  = (blockIdx.x * blockDim.x + threadIdx.x) >> 5;
    const int nw   = (gridDim.x * blockDim.x) >> 5;

    for (int e = wid; e < N_EDGES; e += nw) {
        int s = src[e];
        int d = dst[e];
        int en = e + nw;
        if (en < N_EDGES) {
            int ns = src[en];
            __builtin_prefetch(feature + (size_t)ns * D, 0, 0);  // global_prefetch_b8
        }
        const float4* frow = reinterpret_cast<const float4*>(feature + (size_t)s * D);
        float4 v = frow[lane];                       // lane covers cols 4*lane..4*lane+3
        uint32_t off = ((uint32_t)d * D + (uint32_t)lane * 4u) * 4u;
        gatomic_add_f32(sum, off,       v.x);
        gatomic_add_f32(sum, off + 4u,  v.y);
        gatomic_add_f32(sum, off + 8u,  v.z);
        gatomic_add_f32(sum, off + 12u, v.w);
        if (lane == 0) gatomic_add_f32(deg, (uint32_t)d * 4u, 1.0f);
    }
}

// ---------------------------------------------------------------------------
// Kernel 3: agg = sum / max(deg,1);  out = relu(agg @ W^T) + feature
// One block per 16-row tile (100000/16 = 6250 blocks). 8 waves: wave w owns
// output columns [16w, 16w+16). K=128 -> 32x V_WMMA_F32_16X16X4_F32 per wave.
// W staged transposed in LDS (conflict-free B reads); agg tile staged with
// 132-float pitch (conflict-free A reads).
// ---------------------------------------------------------------------------
__global__ __launch_bounds__(256) void gemm_epilogue_kernel(
    const float* __restrict__ sum, const float* __restrict__ deg,
    const float* __restrict__ W, const float* __restrict__ feature,
    float* __restrict__ out) {
    extern __shared__ float smem[];
    float* lds_wt  = smem;            // [128*128]  wt[k*128 + j] = W[j*128 + k]
    float* lds_agg = smem + D * D;    // [16 * AGG_STRIDE]

    const int tid  = threadIdx.x;
    const int lane = tid & 31;
    const int wave = tid >> 5;                   // 0..7 -> column tile
    const int row0 = blockIdx.x * 16;

    // Stage W transposed (coalesced global read, one-time scattered LDS write).
    for (int idx = tid; idx < D * D; idx += 256) {
        int j = idx >> 7, k = idx & 127;
        lds_wt[k * D + j] = W[idx];
    }
    // Stage agg = sum / max(deg, 1).
    for (int idx = tid; idx < 16 * D; idx += 256) {
        int r = idx >> 7, c = idx & 127;
        float dg  = deg[row0 + r];
        float inv = 1.0f / fmaxf(dg, 1.0f);
        lds_agg[r * AGG_STRIDE + c] = sum[(size_t)(row0 + r) * D + c] * inv;
    }
    __syncthreads();

    // ISA f32 operand layouts (wave32):
    //  A 16x4:  lanes 0-15 hold (M=lane, K=0/1 in v0/v1); lanes 16-31 (M=lane-16, K=2/3)
    //  B 4x16:  lanes 0-15 hold (K=0/1, N=lane);          lanes 16-31 (K=2/3, N=lane-16)
    //  D 16x16: VGPR v holds row v + 8*(lane>=16), col = lane & 15
    const int half = lane >> 4;
    const int l15  = lane & 15;
    const int m    = l15;
    const int n    = wave * 16 + l15;            // global output column
    const int koff = half * 2;

    v8f acc = {};
#pragma unroll
    for (int kk = 0; kk < D / 4; ++kk) {
        int k = kk * 4 + koff;
        v2f a, b;
        a.x = lds_agg[m * AGG_STRIDE + k];
        a.y = lds_agg[m * AGG_STRIDE + k + 1];
        b.x = lds_wt[k * D + n];
        b.y = lds_wt[(k + 1) * D + n];
        acc = __builtin_amdgcn_wmma_f32_16x16x4_f32(
            /*neg_a=*/false, a, /*neg_b=*/false, b,
            /*c_mod=*/(short)0, acc, /*reuse_a=*/false, /*reuse_b=*/false);
    }

    // Epilogue: relu + residual, direct global stores.
#pragma unroll
    for (int v = 0; v < 8; ++v) {
        int r = row0 + v + 8 * half;
        size_t o = (size_t)r * D + n;
        float val = fmaxf(acc[v], 0.0f) + feature[o];
        out[o] = val;
    }
}

// ---------------------------------------------------------------------------
extern "C" void kernel_launch(void* const* d_in, const int* in_sizes, int n_in,
                              void* d_out, int out_size, void* d_ws, size_t ws_size,
                              hipStream_t stream) {
    (void)in_sizes; (void)n_in; (void)out_size; (void)ws_size;
    const float* feature = (const float*)d_in[0];
    const int*   src     = (const int*)d_in[1];
    const int*   dst     = (const int*)d_in[2];
    const float* W       = (const float*)d_in[3];
    float*       out     = (float*)d_out;

    float* sum = (float*)d_ws;                       // N_NODES * D floats
    float* deg = sum + (size_t)N_NODES * D;          // N_NODES floats

    // 1) zero accumulators
    const int n4 = (N_NODES * D + N_NODES) / 4;      // divisible by 4
    zero_ws_kernel<<<(n4 + 255) / 256, 256, 0, stream>>>((float4*)d_ws, n4);

    // 2) edge gather + atomic scatter-add (grid-stride over waves)
    edge_aggregate_kernel<<<2560, 256, 0, stream>>>(feature, src, dst, sum, deg);

    // 3) mean + WMMA GEMM + relu + residual
    const size_t smem_bytes = (size_t)(D * D + 16 * AGG_STRIDE) * sizeof(float); // ~74 KB
    gemm_epilogue_kernel<<<N_NODES / 16, 256, smem_bytes, stream>>>(
        sum, deg, W, feature, out);
}